// ConvAttention_88029649699503
// MI455X (gfx1250) — compile-verified
//
#include <hip/hip_runtime.h>

typedef float v2f __attribute__((ext_vector_type(2)));
typedef float v8f __attribute__((ext_vector_type(8)));

// Shapes: B=2, C=64, H=W=16 (HW=256), L=32
// ws layout (floats):
//   KC    [0, 1048576)        : Kp[b][l][c][h][w], overwritten in place by ck
//   Vp    [1048576, 2097152)  : Vp[b][l][c][h][w]
//   Aperm [2097152, +102400)  : W2 permuted into WMMA A-fragment lane order

// ---------------------------------------------------------------------------
// Kernel 1: K/V projection (Q is provably irrelevant: softmax over j cancels it)
// block = (b, hw), 128 threads (oc 0..63 -> K, 64..127 -> V)
// ---------------------------------------------------------------------------
__global__ __launch_bounds__(128)
void proj_kv_kernel(const float* __restrict__ x, const float* __restrict__ W1,
                    const float* __restrict__ b1, float* __restrict__ KC,
                    float* __restrict__ Vp) {
  __shared__ __align__(16) float xs[64][32];
  const int t   = threadIdx.x;
  const int blk = blockIdx.x;          // b*256 + hw
  const int b   = blk >> 8;
  const int hw  = blk & 255;

  // stage x[b][:, hw, :]  (coalesced: consecutive threads -> consecutive l)
  for (int i = t; i < 2048; i += 128) {
    int c = i >> 5, l = i & 31;
    xs[c][l] = x[((size_t)(b * 64 + c) * 256 + hw) * 32 + l];
  }
  __syncthreads();

  const int oc  = t;            // 0..127
  const int row = 64 + oc;      // W1 rows 64..127 = K, 128..191 = V
  float bias = b1[row];
  float acc[32];
#pragma unroll
  for (int l = 0; l < 32; ++l) acc[l] = bias;

  for (int ci = 0; ci < 64; ++ci) {
    float w = W1[row * 64 + ci];
    const float4* xr = (const float4*)(&xs[ci][0]);
#pragma unroll
    for (int k = 0; k < 8; ++k) {
      float4 v = xr[k];
      acc[4 * k + 0] += w * v.x;
      acc[4 * k + 1] += w * v.y;
      acc[4 * k + 2] += w * v.z;
      acc[4 * k + 3] += w * v.w;
    }
  }

  float* dst = (oc < 64) ? KC : Vp;
  int ch = oc & 63;
  dst += (size_t)b * 524288 + ch * 256 + hw;   // [b][l][ch][hw], l stride 16384
#pragma unroll
  for (int l = 0; l < 32; ++l) dst[l * 16384] = acc[l];
}

// ---------------------------------------------------------------------------
// Kernel 2a: permute W2 into A-fragment order so conv A-loads are coalesced.
// A frag (16x4 f32): lane t, vgpr r -> M = t&15, K = r + 2*(t>>4)
// flat-K order: chunk Q = tile half H (ci/32), q_local: tap = q>>3, cibL=(q&7)*4
// Aperm flat index: tile*25600 + H*12800 + q*64 + t*2 + r
// ---------------------------------------------------------------------------
__global__ __launch_bounds__(256)
void prep_a_kernel(const float* __restrict__ W2, float* __restrict__ Aperm) {
  int e = blockIdx.x * 256 + threadIdx.x;       // 102400 total
  int tile = e / 25600;
  int rem  = e - tile * 25600;
  int H    = rem / 12800;
  int rem2 = rem - H * 12800;
  int q  = rem2 >> 6;
  int tr = rem2 & 63;
  int t = tr >> 1, r = tr & 1;
  int kk  = r + 2 * (t >> 4);
  int ci  = H * 32 + (q & 7) * 4 + kk;
  int tap = q >> 3;
  int co  = tile * 16 + (t & 15);
  Aperm[e] = W2[co * 1600 + ci * 25 + tap];     // W2[co][ci][kh][kw]
}

// ---------------------------------------------------------------------------
// Kernel 2b: 5x5 conv as implicit GEMM on V_WMMA_F32_16X16X4_F32.
// One block per image g = b*32+j; 4 waves = 4 co-tiles of 16.
// N-tile = one pixel row (16 pixels); 16 independent accumulators per wave.
// Input image staged zero-padded (20x20) in LDS, 32 channels per half (50 KB).
// Writes ck in place over KC (safe: image fully staged before stores).
// ---------------------------------------------------------------------------
__global__ __launch_bounds__(128)
void conv_wmma_kernel(float* __restrict__ KC, const float* __restrict__ Aperm,
                      const float* __restrict__ b2) {
  __shared__ float pad[32 * 400];               // 51200 B: [ciL][20][20]
  const int lane = threadIdx.x & 31;
  const int wave = threadIdx.x >> 5;            // co tile 0..3
  const int g    = blockIdx.x;                  // b*32 + j
  const int lo = lane & 15, hi = lane >> 4;
  const int co_base = wave * 16;

  // C/D frag (16x16 f32): lane t, vgpr r -> M = r + 8*(t>>4), N = t&15
  v8f acc[16];
#pragma unroll
  for (int h = 0; h < 16; ++h)
#pragma unroll
    for (int r = 0; r < 8; ++r)
      acc[h][r] = b2[co_base + r + 8 * hi];     // cancels in softmax; exact anyway

  const float* img = KC + (size_t)g * 16384;    // [ci][256]

  for (int H = 0; H < 2; ++H) {
    __syncthreads();                            // waves done reading previous half
    for (int i = threadIdx.x; i < 12800; i += 128) {
      int ciL = i / 400;
      int rem = i - ciL * 400;
      int ph = rem / 20, pw = rem - ph * 20;
      int h = ph - 2, w = pw - 2;
      float v = 0.0f;
      if (h >= 0 && h < 16 && w >= 0 && w < 16)
        v = img[(H * 32 + ciL) * 256 + h * 16 + w];
      pad[i] = v;
    }
    __syncthreads();

    const float* aptr = Aperm + (size_t)((wave * 2 + H) * 200) * 64 + lane * 2;
    for (int q = 0; q < 200; ++q) {
      int tap  = q >> 3;                        // uniform -> scalar
      int cibL = (q & 7) * 4;
      int kh = tap / 5, kw = tap - kh * 5;
      float2 a = *(const float2*)(aptr + q * 64);   // coalesced A fragment
      v2f afrag; afrag[0] = a.x; afrag[1] = a.y;
      // B frag (4x16 f32): lane t, vgpr r -> N = t&15, K = r + 2*(t>>4)
      int ci0   = cibL + 2 * hi;
      int boff0 = ci0 * 400 + kh * 20 + kw + lo;
      int boff1 = boff0 + 400;
#pragma unroll
      for (int h = 0; h < 16; ++h) {
        v2f bfrag;
        bfrag[0] = pad[boff0 + 20 * h];
        bfrag[1] = pad[boff1 + 20 * h];
        acc[h] = __builtin_amdgcn_wmma_f32_16x16x4_f32(
            false, afrag, false, bfrag, (short)0, acc[h], false, false);
      }
    }
  }

  // store ck over the image (in place)
#pragma unroll
  for (int h = 0; h < 16; ++h)
#pragma unroll
    for (int r = 0; r < 8; ++r)
      KC[(size_t)g * 16384 + (co_base + r + 8 * hi) * 256 + h * 16 + lo] =
          acc[h][r];
}

// ---------------------------------------------------------------------------
// Kernel 3: softmax over j, weighted V sum, broadcast over l.
// One thread per (b,c,hw); output row of 32 floats is contiguous.
// ---------------------------------------------------------------------------
__global__ __launch_bounds__(256)
void softmax_av_kernel(const float* __restrict__ KC, const float* __restrict__ Vp,
                       float* __restrict__ out) {
  int idx = blockIdx.x * 256 + threadIdx.x;     // [0, 32768) = b*16384 + c*256 + hw
  int b = idx >> 14;
  int rest = idx & 16383;
  const float* ck = KC + (size_t)b * 524288 + rest;
  const float* vv = Vp + (size_t)b * 524288 + rest;
  float e[32];
  float m = -3.402823466e38f;
#pragma unroll
  for (int j = 0; j < 32; ++j) { e[j] = ck[j * 16384]; m = fmaxf(m, e[j]); }
  float s = 0.0f, o = 0.0f;
#pragma unroll
  for (int j = 0; j < 32; ++j) {
    float p = expf(e[j] - m);
    s += p;
    o += p * vv[j * 16384];
  }
  o /= s;
  float4 o4 = make_float4(o, o, o, o);
  float4* op = (float4*)(out + (size_t)idx * 32);
#pragma unroll
  for (int k = 0; k < 8; ++k) op[k] = o4;
}

// ---------------------------------------------------------------------------
extern "C" void kernel_launch(void* const* d_in, const int* in_sizes, int n_in,
                              void* d_out, int out_size, void* d_ws, size_t ws_size,
                              hipStream_t stream) {
  const float* x  = (const float*)d_in[0];
  const float* W1 = (const float*)d_in[1];
  const float* b1 = (const float*)d_in[2];
  const float* W2 = (const float*)d_in[3];
  const float* b2 = (const float*)d_in[4];
  float* out = (float*)d_out;
  float* ws  = (float*)d_ws;
  float* KC = ws;                 // 1,048,576 floats
  float* Vp = ws + 1048576;       // 1,048,576 floats
  float* Ap = ws + 2097152;       //   102,400 floats (~8.4 MB total)

  prep_a_kernel<<<400, 256, 0, stream>>>(W2, Ap);
  proj_kv_kernel<<<512, 128, 0, stream>>>(x, W1, b1, KC, Vp);
  conv_wmma_kernel<<<64, 128, 0, stream>>>(KC, Ap, b2);
  softmax_av_kernel<<<128, 256, 0, stream>>>(KC, Vp, out);
}